// NMSLambda_10995116277796
// MI455X (gfx1250) — compile-verified
//
#include <hip/hip_runtime.h>
#include <hip/hip_bf16.h>

#define NB   16
#define NBOX 25200
#define NCH  85
#define NCLS 80
#define K    1024
#define MAXDET 300
#define CONF_T 0.3f
#define IOU_T  0.3f
#define MAX_WH 4096.0f
#define ROWS_PER_BLK 128

typedef unsigned int uint;
typedef uint v4u __attribute__((ext_vector_type(4)));
typedef int  v8i __attribute__((ext_vector_type(8)));
typedef int  v4i __attribute__((ext_vector_type(4)));

// ---------------------------------------------------------------------------
// Kernel 1: streaming score/class reduction. TDM stages a 128x85 f32 tile
// into LDS (bulk DMA; rows are 340B so per-lane loads would be misaligned),
// then each thread reduces one row.  Memory-bound: 137MB @ 23.3TB/s ~ 6us.
// ---------------------------------------------------------------------------
__global__ __launch_bounds__(ROWS_PER_BLK)
void k_score(const float* __restrict__ pred,
             float* __restrict__ scores, int* __restrict__ cls)
{
    __shared__ float tile[ROWS_PER_BLK * NCH];
    const int t = threadIdx.x;
    const long long row0 = (long long)blockIdx.x * ROWS_PER_BLK;

    if (t < 32) {  // one wave issues the tensor DMA
        unsigned long long ga = (unsigned long long)(const void*)(pred + row0 * NCH);
        uint ldsoff = (uint)(unsigned long long)(void*)tile;  // low 32 bits = LDS offset
        // D# group 0: count=1 | lds_addr | global_addr[56:0] | type=2
        v4u g0 = { 1u,
                   ldsoff,
                   (uint)(ga & 0xFFFFFFFFull),
                   (uint)((ga >> 32) & 0x1FFFFFFull) | (2u << 30) };
        // D# group 1: data_size=4B; tensor 85 x 128; tile 85 x 128; stride0 = 85
        v8i g1 = { (int)(2u << 16),            // data_size=2 (4 bytes)
                   (int)(85u << 16),           // tensor_dim0[15:0] @ bits[63:48]
                   (int)(128u << 16),          // tensor_dim1[15:0] @ bits[95:80]
                   (int)(85u << 16),           // tile_dim0 @ bits[127:112]
                   (int)128,                   // tile_dim1 @ bits[143:128]
                   (int)85,                    // tensor_dim0_stride[31:0]
                   0, 0 };
        v4i z4 = { 0, 0, 0, 0 };
        v8i z8 = { 0, 0, 0, 0, 0, 0, 0, 0 };
        __builtin_amdgcn_tensor_load_to_lds(g0, g1, z4, z4, z8, 0);
        __builtin_amdgcn_s_wait_tensorcnt(0);
    }
    __syncthreads();

    const float* r = &tile[t * NCH];
    float obj = r[4];
    float m = r[5];
    int   mi = 0;
    for (int k = 1; k < NCLS; ++k) {
        float v = r[5 + k];
        if (v > m) { m = v; mi = k; }
    }
    float conf = obj * m;
    bool valid = (obj > CONF_T) && (conf > CONF_T);
    long long g = row0 + t;
    scores[g] = valid ? conf : -1.0f;
    cls[g] = mi;
}

// ---------------------------------------------------------------------------
// Kernel 2: per-batch top-1024 via 3-level radix-select on float key bits,
// then bitonic sort of <=2048 candidates in LDS.
// ---------------------------------------------------------------------------
__global__ __launch_bounds__(1024)
void k_topk(const float* __restrict__ scores, const int* __restrict__ cls,
            float* __restrict__ tk_score, int* __restrict__ tk_idx,
            int* __restrict__ tk_cls)
{
    const int b = blockIdx.x, t = threadIdx.x;
    const float* sc = scores + (long long)b * NBOX;
    __shared__ uint hist[256];
    __shared__ uint skey[2048];
    __shared__ int  sidx[2048];
    __shared__ uint s_prefix, s_rank, s_cnt;

    if (t == 0) { s_prefix = 0u; s_rank = (uint)K; s_cnt = 0u; }
    __syncthreads();

    for (int lvl = 0; lvl < 3; ++lvl) {
        int shift = 24 - 8 * lvl;
        if (t < 256) hist[t] = 0u;
        __syncthreads();
        uint prefix = s_prefix;
        uint himask = (lvl == 0) ? 0u : (0xFFFFFFFFu << (shift + 8));
        for (int n = t; n < NBOX; n += 1024) {
            float s = sc[n];
            uint key = (s > 0.0f) ? __float_as_uint(s) : 0u;
            if ((key & himask) == (prefix & himask))
                atomicAdd(&hist[(key >> shift) & 0xFF], 1u);
        }
        __syncthreads();
        if (t == 0) {
            uint cum = 0, rank = s_rank;
            for (int bin = 255; bin >= 0; --bin) {
                uint c = hist[bin];
                if (cum + c >= rank) {
                    s_prefix = prefix | ((uint)bin << shift);
                    s_rank = rank - cum;
                    break;
                }
                cum += c;
            }
        }
        __syncthreads();
    }

    uint thr24 = s_prefix >> 8;
    for (int i = t; i < 2048; i += 1024) { skey[i] = 0u; sidx[i] = 0; }
    __syncthreads();
    for (int n = t; n < NBOX; n += 1024) {
        float s = sc[n];
        uint key = (s > 0.0f) ? __float_as_uint(s) : 0u;
        if ((key >> 8) >= thr24) {
            uint w = atomicAdd(&s_cnt, 1u);
            if (w < 2048u) { skey[w] = key; sidx[w] = n; }
        }
    }
    __syncthreads();

    // bitonic sort, descending by key, ascending idx on ties (stable-like)
    for (int kk = 2; kk <= 2048; kk <<= 1) {
        for (int j = kk >> 1; j > 0; j >>= 1) {
            for (int i = t; i < 2048; i += 1024) {
                int ixj = i ^ j;
                if (ixj > i) {
                    uint a = skey[i], c = skey[ixj];
                    int  ia = sidx[i], ic = sidx[ixj];
                    bool aFirst  = (a > c) || (a == c && ia <= ic);
                    bool descSeg = ((i & kk) == 0);
                    bool inOrder = descSeg ? aFirst : !aFirst;
                    if (!inOrder) { skey[i] = c; skey[ixj] = a; sidx[i] = ic; sidx[ixj] = ia; }
                }
            }
            __syncthreads();
        }
    }

    uint key = skey[t];
    int  idx = sidx[t];
    long long o = (long long)b * K + t;
    tk_score[o] = (key != 0u) ? __uint_as_float(key) : -1.0f;
    tk_idx[o]   = idx;
    tk_cls[o]   = cls[(long long)b * NBOX + idx];
}

// ---------------------------------------------------------------------------
// Kernel 3: IoU bitmask rows -> L2-resident workspace, wave-0 serial greedy
// suppression (word-parallel over 32 lanes), block scan, top-300 emit.
// ---------------------------------------------------------------------------
__global__ __launch_bounds__(1024)
void k_nms(const float* __restrict__ pred,
           const float* __restrict__ tk_score, const int* __restrict__ tk_idx,
           const int* __restrict__ tk_cls,
           uint* __restrict__ gmask, float* __restrict__ out)
{
    const int b = blockIdx.x, t = threadIdx.x;
    __shared__ float sx1[K], sy1[K], sx2[K], sy2[K], ssc[K], scl[K];
    __shared__ uint ssup[K / 32];
    __shared__ int  sscan[K];

    long long o = (long long)b * K + t;
    float s = tk_score[o];
    int idx = tk_idx[o];
    float c = (float)tk_cls[o];
    const float* p = pred + ((long long)b * NBOX + idx) * NCH;
    float x = p[0], y = p[1], w = p[2], h = p[3];
    float off = c * MAX_WH;
    sx1[t] = x - 0.5f * w + off;  sy1[t] = y - 0.5f * h + off;
    sx2[t] = x + 0.5f * w + off;  sy2[t] = y + 0.5f * h + off;
    ssc[t] = s;  scl[t] = c;
    __syncthreads();

    float ax1 = sx1[t], ay1 = sy1[t], ax2 = sx2[t], ay2 = sy2[t];
    float aarea = fmaxf(ax2 - ax1, 0.0f) * fmaxf(ay2 - ay1, 0.0f);
    uint* myrow = gmask + ((long long)b * K + t) * 32;
    for (int wd = 0; wd < 32; ++wd) {
        uint bits = 0u;
        for (int jj = 0; jj < 32; ++jj) {
            int j = wd * 32 + jj;
            float bx1 = sx1[j], by1 = sy1[j], bx2 = sx2[j], by2 = sy2[j];
            float barea = fmaxf(bx2 - bx1, 0.0f) * fmaxf(by2 - by1, 0.0f);
            float iw = fmaxf(fminf(ax2, bx2) - fmaxf(ax1, bx1), 0.0f);
            float ih = fmaxf(fminf(ay2, by2) - fmaxf(ay1, by1), 0.0f);
            float inter = iw * ih;
            float iou = inter / (aarea + barea - inter + 1e-9f);
            if (j > t && iou > IOU_T) bits |= (1u << jj);
        }
        myrow[wd] = bits;
    }
    __threadfence();
    __syncthreads();

    if (t < 32) {
        uint sup = 0u;
        const uint* rows = gmask + (long long)b * K * 32;
        for (int i = 0; i < K; ++i) {
            uint wsup = __shfl(sup, i >> 5, 32);
            bool supp = (wsup >> (i & 31)) & 1u;
            if (!supp && ssc[i] > 0.0f)
                sup |= rows[(long long)i * 32 + t];
        }
        ssup[t] = sup;
    }
    __syncthreads();

    int keep = (ssc[t] > 0.0f) && !((ssup[t >> 5] >> (t & 31)) & 1u);
    sscan[t] = keep;
    __syncthreads();
    for (int ofn = 1; ofn < K; ofn <<= 1) {
        int add = (t >= ofn) ? sscan[t - ofn] : 0;
        __syncthreads();
        sscan[t] += add;
        __syncthreads();
    }
    int pos = sscan[t] - keep;
    int total = sscan[K - 1];

    float* dets = out;
    float* msk  = out + NB * MAXDET * 6;
    if (keep && pos < MAXDET) {
        long long d = ((long long)b * MAXDET + pos) * 6;
        float offc = scl[t] * MAX_WH;
        dets[d + 0] = sx1[t] - offc;
        dets[d + 1] = sy1[t] - offc;
        dets[d + 2] = sx2[t] - offc;
        dets[d + 3] = sy2[t] - offc;
        dets[d + 4] = ssc[t];
        dets[d + 5] = scl[t];
        msk[(long long)b * MAXDET + pos] = 1.0f;
    }
    int lim = total < MAXDET ? total : MAXDET;
    if (t < MAXDET && t >= lim) {
        long long d = ((long long)b * MAXDET + t) * 6;
        for (int q = 0; q < 6; ++q) dets[d + q] = 0.0f;
        msk[(long long)b * MAXDET + t] = 0.0f;
    }
}

// ---------------------------------------------------------------------------
extern "C" void kernel_launch(void* const* d_in, const int* in_sizes, int n_in,
                              void* d_out, int out_size, void* d_ws, size_t ws_size,
                              hipStream_t stream)
{
    const float* pred = (const float*)d_in[0];
    float* out = (float*)d_out;

    char* ws = (char*)d_ws;
    float* scores   = (float*)(ws);                                   // 16*25200*4
    int*   cls      = (int*)  (ws + 1612800);                         // 16*25200*4
    float* tk_score = (float*)(ws + 3225600);                         // 16*1024*4
    int*   tk_idx   = (int*)  (ws + 3291136);
    int*   tk_cls   = (int*)  (ws + 3356672);
    uint*  gmask    = (uint*) (ws + 3422208);                         // 16*1024*32*4

    int nblk = (NB * NBOX) / ROWS_PER_BLK;  // 403200/128 = 3150 exact
    k_score<<<nblk, ROWS_PER_BLK, 0, stream>>>(pred, scores, cls);
    k_topk <<<NB, 1024, 0, stream>>>(scores, cls, tk_score, tk_idx, tk_cls);
    k_nms  <<<NB, 1024, 0, stream>>>(pred, tk_score, tk_idx, tk_cls, gmask, out);
}